// TopKRouter_29600914604180
// MI455X (gfx1250) — compile-verified
//
#include <hip/hip_runtime.h>
#include <stdint.h>

// CDNA5 (gfx1250) fused MoE top-2 router:
//   logits = x[16384,4096] @ W[64,4096]^T via V_WMMA_F32_16X16X4_F32,
//   double-buffered GLOBAL_LOAD_ASYNC_TO_LDS_B128 staging (ASYNCcnt),
//   fused per-token top-2 + softmax(2) epilogue.

typedef __attribute__((ext_vector_type(2))) float v2f;
typedef __attribute__((ext_vector_type(8))) float v8f;

#define HIDDEN    4096
#define EXPERTS   64
#define MTILE     32                  // tokens per block
#define KCHUNK    64                  // K slice per stage
#define NCHUNK    (HIDDEN / KCHUNK)   // 64
#define TSTRIDE   68                  // row stride (floats): 16B-aligned rows, bank-spread
#define LG_STRIDE 65

__device__ __forceinline__ void async_b128(uint32_t lds, const float* g) {
    // per lane, copies 16 bytes from the global address in g to the LDS address in lds; tracked by ASYNCcnt
    asm volatile("global_load_async_to_lds_b128 %0, %1, off"
                 :: "v"(lds), "v"(g) : "memory");
}

__launch_bounds__(256, 2)
__global__ void topk_router_kernel(const float* __restrict__ x,
                                   const float* __restrict__ W,
                                   float* __restrict__ out_scores,
                                   int* __restrict__ out_idx) {
    __shared__ float xs[2][MTILE   * TSTRIDE];   // 2 x 8.7 KB  A stages (tokens x K)
    __shared__ float ws[2][EXPERTS * TSTRIDE];   // 2 x 17.4 KB B stages (experts x K, row-major copy)
    __shared__ float lg[MTILE * LG_STRIDE];      // 8.3 KB      logits

    const int tid  = threadIdx.x;
    const int lane = tid & 31;
    const int wave = tid >> 5;
    const int M0   = (wave >> 2) * 16;   // 2 row tiles
    const int N0   = (wave & 3) * 16;    // 4 col tiles
    const int half = lane >> 4;          // K-pair select
    const int r    = lane & 15;          // A: row, B: col (lane = N)

    const long tok0 = (long)blockIdx.x * MTILE;

    // ---- cooperative-copy assignment: row = tid>>4, float4 col = tid&15 ----
    const int trow = tid >> 4;           // 0..15
    const int c4   = tid & 15;           // 0..15
    const float* xg0 = x + (tok0 + trow) * HIDDEN + c4 * 4;
    const float* xg1 = xg0 + 16L * HIDDEN;
    const float* wg0 = W + (long)trow * HIDDEN + c4 * 4;   // experts trow, +16, +32, +48

    uint32_t xl0[2], xl1[2], wl0[2];
    #pragma unroll
    for (int p = 0; p < 2; ++p) {
        xl0[p] = (uint32_t)(uintptr_t)&xs[p][ trow       * TSTRIDE + c4 * 4];
        xl1[p] = (uint32_t)(uintptr_t)&xs[p][(trow + 16) * TSTRIDE + c4 * 4];
        wl0[p] = (uint32_t)(uintptr_t)&ws[p][ trow       * TSTRIDE + c4 * 4];
    }

    // ---- prologue: stage chunk 0 into buffer 0 (6 async b128 / thread) ----
    async_b128(xl0[0], xg0);
    async_b128(xl1[0], xg1);
    #pragma unroll
    for (int j = 0; j < 4; ++j)
        async_b128(wl0[0] + (uint32_t)(j * 16 * TSTRIDE * 4), wg0 + (long)j * 16 * HIDDEN);

    v8f acc0 = {}, acc1 = {};

    for (int i = 0; i < NCHUNK; ++i) {
        const int p = i & 1;
        if (i + 1 < NCHUNK) {
            // issue chunk i+1 into the other buffer, then wait for chunk i
            const int kb = (i + 1) * KCHUNK;
            async_b128(xl0[1 - p], xg0 + kb);
            async_b128(xl1[1 - p], xg1 + kb);
            #pragma unroll
            for (int j = 0; j < 4; ++j)
                async_b128(wl0[1 - p] + (uint32_t)(j * 16 * TSTRIDE * 4),
                           wg0 + (long)j * 16 * HIDDEN + kb);
            asm volatile("s_wait_asynccnt 0x6" ::: "memory");  // chunk i landed (loads retire in order)
        } else {
            asm volatile("s_wait_asynccnt 0x0" ::: "memory");  // last chunk landed
        }
        __syncthreads();   // everyone's chunk-i data visible in LDS

        const float* xb = &xs[p][0];
        const float* wb = &ws[p][0];
        // 16 x V_WMMA_F32_16X16X4_F32, two accumulators for ILP
        #pragma unroll
        for (int k0 = 0; k0 < KCHUNK; k0 += 8) {
            v2f a0 = *(const v2f*)(&xb[(M0 + r) * TSTRIDE + k0 + 2 * half]);
            v2f b0 = *(const v2f*)(&wb[(N0 + r) * TSTRIDE + k0 + 2 * half]);
            acc0 = __builtin_amdgcn_wmma_f32_16x16x4_f32(false, a0, false, b0,
                                                         (short)0, acc0, false, false);
            v2f a1 = *(const v2f*)(&xb[(M0 + r) * TSTRIDE + k0 + 4 + 2 * half]);
            v2f b1 = *(const v2f*)(&wb[(N0 + r) * TSTRIDE + k0 + 4 + 2 * half]);
            acc1 = __builtin_amdgcn_wmma_f32_16x16x4_f32(false, a1, false, b1,
                                                         (short)0, acc1, false, false);
        }
        __syncthreads();   // all reads of buf p done before iter i+1 overwrites it
    }

    const v8f acc = acc0 + acc1;

    // ---- C tile -> LDS logits (documented 16x16 f32 C/D layout) ----
    #pragma unroll
    for (int i = 0; i < 8; ++i) {
        const int row = M0 + half * 8 + i;
        const int col = N0 + r;
        lg[row * LG_STRIDE + col] = acc[i];
    }
    __syncthreads();

    // ---- fused top-2 + softmax over the 2 selected logits ----
    if (tid < MTILE) {
        const float* row = &lg[tid * LG_STRIDE];
        float v0 = row[0]; int i0 = 0;
        float v1 = -3.0e38f; int i1 = 0;
        #pragma unroll 8
        for (int e = 1; e < EXPERTS; ++e) {
            const float f = row[e];
            if (f > v0)      { v1 = v0; i1 = i0; v0 = f; i0 = e; }  // strict >: lowest index wins ties
            else if (f > v1) { v1 = f; i1 = e; }
        }
        const float d   = __expf(v1 - v0);        // <= 1, numerically safe
        const float inv = 1.0f / (1.0f + d);
        const long  t   = tok0 + tid;
        out_scores[t * 2 + 0] = inv;
        out_scores[t * 2 + 1] = d * inv;
        out_idx   [t * 2 + 0] = i0;
        out_idx   [t * 2 + 1] = i1;
    }
}

extern "C" void kernel_launch(void* const* d_in, const int* in_sizes, int n_in,
                              void* d_out, int out_size, void* d_ws, size_t ws_size,
                              hipStream_t stream) {
    const float* x = (const float*)d_in[0];
    const float* W = (const float*)d_in[1];
    const int tokens = in_sizes[0] / HIDDEN;              // 16384
    float* out_scores = (float*)d_out;                    // 2*tokens fp32 scores
    int*   out_idx    = (int*)d_out + (size_t)tokens * 2; // then 2*tokens int32 indices
    const int grid = tokens / MTILE;                      // 512 blocks
    topk_router_kernel<<<grid, 256, 0, stream>>>(x, W, out_scores, out_idx);
}